// GraphTransformerNeighbors_2104533975110
// MI455X (gfx1250) — compile-verified
//
#include <hip/hip_runtime.h>
#include <math.h>

#define IN_CH 128
#define HIDH  128   // HEADS*HID
#define OUTC  64

#define USE_ASYNC_LDS 1

typedef __attribute__((ext_vector_type(2))) float v2f;
typedef __attribute__((ext_vector_type(8))) float v8f;

__device__ __forceinline__ v8f wmma_f32(v2f a, v2f b, v8f c) {
    return __builtin_amdgcn_wmma_f32_16x16x4_f32(
        /*neg_a=*/false, a, /*neg_b=*/false, b,
        /*c_mod=*/(short)0, c, /*reuse_a=*/false, /*reuse_b=*/false);
}

// Stage one dword from global into LDS. On gfx1250 use the async LDS-direct
// path (GLOBAL_LOAD_ASYNC_TO_LDS_B32, tracked by ASYNCcnt) — per-lane LDS
// byte offset is the low 32 bits of the generic pointer (ISA: LDS aperture
// uses addr[31:0] directly).
__device__ __forceinline__ void stage_b32(const float* gsrc, float* lds_dst) {
#if USE_ASYNC_LDS
    unsigned loff = (unsigned)(size_t)lds_dst;
    asm volatile("global_load_async_to_lds_b32 %0, %1, off"
                 :: "v"(loff), "v"(gsrc) : "memory");
#else
    *lds_dst = *gsrc;
#endif
}
__device__ __forceinline__ void stage_wait() {
#if USE_ASYNC_LDS
    asm volatile("s_wait_asynccnt 0x0" ::: "memory");
#endif
}

__device__ __forceinline__ void atomicMaxF(float* addr, float val) {
    // IEEE-754 monotonic-bits trick; init value must be -inf (0xFF800000).
    if (val >= 0.0f) atomicMax((int*)addr, __float_as_int(val));
    else             atomicMin((unsigned int*)addr, __float_as_uint(val));
}

__global__ void fill_kernel(float* p, float val, int n) {
    int i = blockIdx.x * blockDim.x + threadIdx.x;
    int stride = gridDim.x * blockDim.x;
    for (; i < n; i += stride) p[i] = val;
}

// ---------------------------------------------------------------------------
// C[M x NC] = A[M x K] * W[K x NC] (+bias). Compile-time K, NC.
// Block = 128 threads = 4 waves; all 4 waves share one 16-wide column panel
// of W, staged once into LDS (async loads). Each wave owns one 16x16 C tile.
// f32 WMMA layouts (wave32): A: lane&15 = M row, K = vgpr + 2*(lane>>4);
// B: N = lane&15, K = vgpr + 2*(lane>>4); C: M = vgpr + 8*(lane>>4), N = lane&15.
// Out-of-range M rows are clamped on load (they only affect unstored C rows).
// ---------------------------------------------------------------------------
template<int K, int NC>
__global__ void gemm_wmma_f32(const float* __restrict__ A,
                              const float* __restrict__ W,
                              const float* __restrict__ bias,
                              float* __restrict__ C, int M) {
    __shared__ float bp[K * 16];
    const int tilesM = (M + 15) >> 4;
    const int tmB = (tilesM + 3) >> 2;          // M-tile blocks per column
    const int tn = blockIdx.x / tmB;
    const int tmblk = blockIdx.x % tmB;
    const int col0 = tn << 4;

    // Stage B panel (K x 16) cooperatively.
    for (int i = threadIdx.x; i < K * 16; i += 128) {
        int kk = i >> 4, n = i & 15;
        stage_b32(W + (size_t)kk * NC + col0 + n, &bp[i]);
    }
    stage_wait();
    __syncthreads();

    const int wv = threadIdx.x >> 5, lane = threadIdx.x & 31;
    const int tm = tmblk * 4 + wv;
    if (tm >= tilesM) return;
    const int row0 = tm << 4;
    const int half = lane >> 4, l15 = lane & 15;
    int arow = row0 + l15;
    if (arow >= M) arow = M - 1;                // clamp: garbage rows unstored
    const float* ap = A + (size_t)arow * K + 2 * half;

    v8f acc = {};
    #pragma unroll 4
    for (int k0 = 0; k0 < K; k0 += 4) {
        v2f a; a[0] = ap[k0]; a[1] = ap[k0 + 1];
        int bi = (k0 + 2 * half) * 16 + l15;
        v2f b; b[0] = bp[bi]; b[1] = bp[bi + 16];
        acc = wmma_f32(a, b, acc);
    }
    float bv = (bias != nullptr) ? bias[col0 + l15] : 0.0f;
    #pragma unroll
    for (int v = 0; v < 8; ++v) {
        int r = row0 + v + 8 * half;
        if (r < M) C[(size_t)r * NC + col0 + l15] = acc[v] + bv;
    }
}

// ---------------------------------------------------------------------------
// Fused q/k/v/skip: four [M x 128] x [128 x 64] GEMMs sharing A. Each A
// fragment feeds 4 WMMAs (4 accumulators, 4 B panels in 32KB LDS).
// ---------------------------------------------------------------------------
__global__ void gemm4_wmma_f32(const float* __restrict__ A,
                               const float* __restrict__ W0, const float* __restrict__ W1,
                               const float* __restrict__ W2, const float* __restrict__ W3,
                               const float* __restrict__ b0, const float* __restrict__ b1,
                               const float* __restrict__ b2, const float* __restrict__ b3,
                               float* __restrict__ O0, float* __restrict__ O1,
                               float* __restrict__ O2, float* __restrict__ O3, int M) {
    constexpr int K = 128, NC = 64;
    __shared__ float bp[4 * K * 16];            // 32 KB
    const int tilesM = (M + 15) >> 4;
    const int tmB = (tilesM + 3) >> 2;
    const int tn = blockIdx.x / tmB;            // 0..3
    const int tmblk = blockIdx.x % tmB;
    const int col0 = tn << 4;

    #pragma unroll
    for (int mat = 0; mat < 4; ++mat) {
        const float* Wm = (mat == 0) ? W0 : (mat == 1) ? W1 : (mat == 2) ? W2 : W3;
        for (int i = threadIdx.x; i < K * 16; i += 128) {
            int kk = i >> 4, n = i & 15;
            stage_b32(Wm + (size_t)kk * NC + col0 + n, &bp[mat * K * 16 + i]);
        }
    }
    stage_wait();
    __syncthreads();

    const int wv = threadIdx.x >> 5, lane = threadIdx.x & 31;
    const int tm = tmblk * 4 + wv;
    if (tm >= tilesM) return;
    const int row0 = tm << 4;
    const int half = lane >> 4, l15 = lane & 15;
    int arow = row0 + l15;
    if (arow >= M) arow = M - 1;
    const float* ap = A + (size_t)arow * K + 2 * half;

    v8f ac0 = {}, ac1 = {}, ac2 = {}, ac3 = {};
    #pragma unroll 4
    for (int k0 = 0; k0 < K; k0 += 4) {
        v2f a; a[0] = ap[k0]; a[1] = ap[k0 + 1];
        int bi = (k0 + 2 * half) * 16 + l15;
        v2f b;
        b[0] = bp[bi];              b[1] = bp[bi + 16];              ac0 = wmma_f32(a, b, ac0);
        b[0] = bp[2048 + bi];       b[1] = bp[2048 + bi + 16];       ac1 = wmma_f32(a, b, ac1);
        b[0] = bp[4096 + bi];       b[1] = bp[4096 + bi + 16];       ac2 = wmma_f32(a, b, ac2);
        b[0] = bp[6144 + bi];       b[1] = bp[6144 + bi + 16];       ac3 = wmma_f32(a, b, ac3);
    }
    float bv0 = b0[col0 + l15], bv1 = b1[col0 + l15];
    float bv2 = b2[col0 + l15], bv3 = b3[col0 + l15];
    #pragma unroll
    for (int v = 0; v < 8; ++v) {
        int r = row0 + v + 8 * half;
        if (r < M) {
            size_t o = (size_t)r * NC + col0 + l15;
            O0[o] = ac0[v] + bv0;
            O1[o] = ac1[v] + bv1;
            O2[o] = ac2[v] + bv2;
            O3[o] = ac3[v] + bv3;
        }
    }
}

// a_src[n][hd] = sum_c h[n][hd][c]*att_src[hd][c]; likewise a_dst.
__global__ void gat_attn_coef(const float* __restrict__ h,
                              const float* __restrict__ att_src,
                              const float* __restrict__ att_dst,
                              float* __restrict__ a_src,
                              float* __restrict__ a_dst, int Nn) {
    int t = blockIdx.x * blockDim.x + threadIdx.x;
    if (t >= Nn * 2) return;
    int n = t >> 1, hd = t & 1;
    const float* hp = h + (size_t)n * HIDH + hd * 64;
    const float* as = att_src + hd * 64;
    const float* ad = att_dst + hd * 64;
    float s = 0.0f, d = 0.0f;
    for (int c = 0; c < 64; ++c) {
        float hv = hp[c];
        s += hv * as[c];
        d += hv * ad[c];
    }
    a_src[t] = s;
    a_dst[t] = d;
}

// Messages 0..E-1 are edges; E..E+N-1 are self loops.
__global__ void gat_edge_max(const int* __restrict__ ei, int E_, int Nn,
                             const float* __restrict__ a_src,
                             const float* __restrict__ a_dst,
                             float* __restrict__ e1, float* __restrict__ m1) {
    int idx = blockIdx.x * blockDim.x + threadIdx.x;
    int total = E_ + Nn;
    if (idx >= total) return;
    int s, d;
    if (idx < E_) { s = ei[idx]; d = ei[E_ + idx]; }
    else          { s = d = idx - E_; }
    for (int hd = 0; hd < 2; ++hd) {
        float t = a_src[2 * s + hd] + a_dst[2 * d + hd];
        float e = (t > 0.0f) ? t : 0.2f * t;   // leaky_relu(0.2)
        e1[2 * idx + hd] = e;
        atomicMaxF(&m1[2 * d + hd], e);
    }
}

// One wave per message: accumulate ex * h[src] into out1[dst], ex into s1[dst].
__global__ void gat_edge_acc(const int* __restrict__ ei, int E_, int Nn,
                             const float* __restrict__ e1,
                             const float* __restrict__ m1,
                             const float* __restrict__ h,
                             float* __restrict__ s1, float* __restrict__ out1) {
    int wave = blockIdx.x * (blockDim.x >> 5) + (threadIdx.x >> 5);
    int lane = threadIdx.x & 31;
    int total = E_ + Nn;
    if (wave >= total) return;
    int s, d;
    if (wave < E_) { s = ei[wave]; d = ei[E_ + wave]; }
    else           { s = d = wave - E_; }
    int c0 = lane * 4;          // 4 channels per lane over 128
    int hd = c0 >> 6;           // head 0 for lanes 0-15, head 1 for 16-31
    float ex = expf(e1[2 * wave + hd] - m1[2 * d + hd]);
    if (lane == 0 || lane == 16) atomicAdd(&s1[2 * d + hd], ex);
    const float4 hv = *(const float4*)(h + (size_t)s * HIDH + c0);
    float* op = out1 + (size_t)d * HIDH + c0;
    atomicAdd(op + 0, ex * hv.x);
    atomicAdd(op + 1, ex * hv.y);
    atomicAdd(op + 2, ex * hv.z);
    atomicAdd(op + 3, ex * hv.w);
}

__global__ void gat_finalize(const float* __restrict__ out1,
                             const float* __restrict__ s1,
                             const float* __restrict__ b1,
                             float* __restrict__ x1, int Nn) {
    int i = blockIdx.x * blockDim.x + threadIdx.x;
    if (i >= Nn * HIDH) return;
    int n = i >> 7, c = i & 127, hd = c >> 6;
    float val = out1[i] / s1[2 * n + hd] + b1[c];   // s1 > 0 (self-loops)
    x1[i] = (val > 0.0f) ? val : 0.0f;
}

// One wave per edge: score = dot(q[dst], k[src]) / sqrt(64); store + max.
__global__ void tr_edge_max(const int* __restrict__ ei, int E_,
                            const float* __restrict__ q,
                            const float* __restrict__ k,
                            float* __restrict__ e2, float* __restrict__ m2) {
    int wave = blockIdx.x * (blockDim.x >> 5) + (threadIdx.x >> 5);
    int lane = threadIdx.x & 31;
    if (wave >= E_) return;
    int s = ei[wave], d = ei[E_ + wave];
    int c = lane * 2;
    const float2 qv = *(const float2*)(q + (size_t)d * OUTC + c);
    const float2 kv = *(const float2*)(k + (size_t)s * OUTC + c);
    float p = qv.x * kv.x + qv.y * kv.y;
    for (int m = 16; m > 0; m >>= 1) p += __shfl_xor(p, m, 32);
    p *= 0.125f;  // 1/sqrt(64)
    if (lane == 0) {
        e2[wave] = p;
        atomicMaxF(&m2[d], p);
    }
}

__global__ void tr_edge_acc(const int* __restrict__ ei, int E_,
                            const float* __restrict__ e2,
                            const float* __restrict__ m2,
                            const float* __restrict__ v,
                            float* __restrict__ s2, float* __restrict__ agg) {
    int wave = blockIdx.x * (blockDim.x >> 5) + (threadIdx.x >> 5);
    int lane = threadIdx.x & 31;
    if (wave >= E_) return;
    int s = ei[wave], d = ei[E_ + wave];
    float ex = expf(e2[wave] - m2[d]);
    if (lane == 0) atomicAdd(&s2[d], ex);
    int c = lane * 2;
    const float2 vv = *(const float2*)(v + (size_t)s * OUTC + c);
    float* ap = agg + (size_t)d * OUTC + c;
    atomicAdd(ap + 0, ex * vv.x);
    atomicAdd(ap + 1, ex * vv.y);
}

// One wave per node: out = log_softmax(agg/s2 + skip) over 64 channels.
__global__ void final_logsoftmax(const float* __restrict__ agg,
                                 const float* __restrict__ s2,
                                 const float* __restrict__ skip,
                                 float* __restrict__ out, int Nn) {
    int node = blockIdx.x * (blockDim.x >> 5) + (threadIdx.x >> 5);
    int lane = threadIdx.x & 31;
    if (node >= Nn) return;
    float sv = s2[node];
    float inv = (sv > 0.0f) ? (1.0f / sv) : 0.0f;  // nodes with no in-edges
    int c = lane * 2;
    const float2 av = *(const float2*)(agg + (size_t)node * OUTC + c);
    const float2 sk = *(const float2*)(skip + (size_t)node * OUTC + c);
    float v0 = av.x * inv + sk.x;
    float v1 = av.y * inv + sk.y;
    float mx = fmaxf(v0, v1);
    for (int m = 16; m > 0; m >>= 1) mx = fmaxf(mx, __shfl_xor(mx, m, 32));
    float se = expf(v0 - mx) + expf(v1 - mx);
    for (int m = 16; m > 0; m >>= 1) se += __shfl_xor(se, m, 32);
    float ls = logf(se);
    out[(size_t)node * OUTC + c]     = v0 - mx - ls;
    out[(size_t)node * OUTC + c + 1] = v1 - mx - ls;
}

extern "C" void kernel_launch(void* const* d_in, const int* in_sizes, int n_in,
                              void* d_out, int out_size, void* d_ws, size_t ws_size,
                              hipStream_t stream) {
    const float* x       = (const float*)d_in[0];
    const int*   ei      = (const int*)d_in[1];
    const float* W1      = (const float*)d_in[2];
    const float* att_src = (const float*)d_in[3];
    const float* att_dst = (const float*)d_in[4];
    const float* b1      = (const float*)d_in[5];
    const float* Wq      = (const float*)d_in[6];
    const float* bq      = (const float*)d_in[7];
    const float* Wk      = (const float*)d_in[8];
    const float* bk      = (const float*)d_in[9];
    const float* Wv      = (const float*)d_in[10];
    const float* bv      = (const float*)d_in[11];
    const float* Wskip   = (const float*)d_in[12];
    const float* bskip   = (const float*)d_in[13];
    float* out = (float*)d_out;

    const int Nn = in_sizes[0] / IN_CH;
    const int E_ = in_sizes[1] / 2;
    const int TOT = E_ + Nn;

    // Workspace layout (fp32)
    float* ws   = (float*)d_ws;
    float* h    = ws;              ws += (size_t)Nn * HIDH;   // N x 128
    float* asr  = ws;              ws += (size_t)Nn * 2;
    float* ads  = ws;              ws += (size_t)Nn * 2;
    float* e1   = ws;              ws += (size_t)TOT * 2;
    float* m1   = ws;              ws += (size_t)Nn * 2;
    float* s1   = ws;              ws += (size_t)Nn * 2;
    float* out1 = ws;              ws += (size_t)Nn * HIDH;
    float* x1   = ws;              ws += (size_t)Nn * HIDH;
    float* q    = ws;              ws += (size_t)Nn * OUTC;
    float* k    = ws;              ws += (size_t)Nn * OUTC;
    float* v    = ws;              ws += (size_t)Nn * OUTC;
    float* skip = ws;              ws += (size_t)Nn * OUTC;
    float* e2   = ws;              ws += (size_t)E_;
    float* m2   = ws;              ws += (size_t)Nn;
    float* s2   = ws;              ws += (size_t)Nn;
    float* agg  = ws;              ws += (size_t)Nn * OUTC;

    const int tilesM = (Nn + 15) / 16;
    const int tmB = (tilesM + 3) / 4;

    // 1. h = x @ W1 (WMMA f32, B panel staged via async LDS loads)
    gemm_wmma_f32<128, 128><<<8 * tmB, 128, 0, stream>>>(x, W1, nullptr, h, Nn);
    // 2. attention coefficients
    gat_attn_coef<<<(Nn * 2 + 255) / 256, 256, 0, stream>>>(h, att_src, att_dst, asr, ads, Nn);
    // 3. init accumulators (every call: harness replays without re-poisoning)
    fill_kernel<<<512, 256, 0, stream>>>(m1, -INFINITY, Nn * 2);
    fill_kernel<<<512, 256, 0, stream>>>(s1, 0.0f, Nn * 2);
    fill_kernel<<<1024, 256, 0, stream>>>(out1, 0.0f, Nn * HIDH);
    fill_kernel<<<512, 256, 0, stream>>>(m2, -INFINITY, Nn);
    fill_kernel<<<512, 256, 0, stream>>>(s2, 0.0f, Nn);
    fill_kernel<<<1024, 256, 0, stream>>>(agg, 0.0f, Nn * OUTC);
    // 4. GAT segment-max pass
    gat_edge_max<<<(TOT + 255) / 256, 256, 0, stream>>>(ei, E_, Nn, asr, ads, e1, m1);
    // 5. GAT exp/accumulate pass (wave per message)
    gat_edge_acc<<<(TOT + 7) / 8, 256, 0, stream>>>(ei, E_, Nn, e1, m1, h, s1, out1);
    // 6. x1 = relu(out1/s1 + b1)
    gat_finalize<<<(Nn * HIDH + 255) / 256, 256, 0, stream>>>(out1, s1, b1, x1, Nn);
    // 7. fused q/k/v/skip GEMMs (one A load feeds 4 WMMAs)
    gemm4_wmma_f32<<<4 * tmB, 128, 0, stream>>>(x1, Wq, Wk, Wv, Wskip,
                                                bq, bk, bv, bskip,
                                                q, k, v, skip, Nn);
    // 8. Transformer segment-max pass (wave per edge)
    tr_edge_max<<<(E_ + 7) / 8, 256, 0, stream>>>(ei, E_, q, k, e2, m2);
    // 9. Transformer exp/accumulate pass
    tr_edge_acc<<<(E_ + 7) / 8, 256, 0, stream>>>(ei, E_, e2, m2, v, s2, agg);
    // 10. out = log_softmax(agg/s2 + skip)
    final_logsoftmax<<<(Nn + 7) / 8, 256, 0, stream>>>(agg, s2, skip, out, Nn);
}